// hypergraph_part_40218073760239
// MI455X (gfx1250) — compile-verified
//
#include <hip/hip_runtime.h>
#include <math.h>

#define DN   1024
#define MN   512
#define NIN  128
#define NOUT 128
#define NCOMB (DN + MN)   // 1536

typedef float v2f __attribute__((ext_vector_type(2)));
typedef float v8f __attribute__((ext_vector_type(8)));

__device__ __forceinline__ float lrelu02(float x) { return x > 0.0f ? x : 0.2f * x; }

// ---------------------------------------------------------------------------
// K1: gather comb = [c_embeddings[c_it]; m_embeddings[medicine_it]]  (1536x128)
// ---------------------------------------------------------------------------
__global__ void k_gather(const int* __restrict__ c_it, const int* __restrict__ m_it,
                         const float* __restrict__ cemb, const float* __restrict__ memb,
                         float* __restrict__ comb) {
    int i = blockIdx.x;      // 0..1535
    int t = threadIdx.x;     // 0..127
    float v;
    if (i < DN) v = cemb[(long)c_it[i] * NIN + t];
    else        v = memb[(long)m_it[i - DN] * NIN + t];
    comb[i * NIN + t] = v;
}

// ---------------------------------------------------------------------------
// K2: xg = comb @ W_gat  via V_WMMA_F32_16X16X4_F32 (one wave per 16x16 tile)
//   grid = 96 blocks x 256 threads (8 waves/block); 96 M-tiles x 8 N-tiles
// ---------------------------------------------------------------------------
__global__ __launch_bounds__(256) void k_gemm_wmma(const float* __restrict__ A,
                                                   const float* __restrict__ B,
                                                   float* __restrict__ C) {
    int lane = threadIdx.x & 31;
    int wave = threadIdx.x >> 5;
    int tile = blockIdx.x * 8 + wave;     // 0..767
    int tm = tile >> 3;                   // 0..95
    int tn = tile & 7;                    // 0..7
    int mrow = tm * 16 + (lane & 15);     // A: lane%16 = M
    int ncol = tn * 16 + (lane & 15);     // B/C/D: lane%16 = N
    int kh   = (lane >> 4) * 2;           // lane-group picks K pair {0,1} or {2,3}

    v8f acc = {};
    for (int k0 = 0; k0 < NIN; k0 += 4) {
        v2f av, bv;
        av.x = A[mrow * NIN + k0 + kh];
        av.y = A[mrow * NIN + k0 + kh + 1];
        bv.x = B[(k0 + kh)     * NOUT + ncol];
        bv.y = B[(k0 + kh + 1) * NOUT + ncol];
        acc = __builtin_amdgcn_wmma_f32_16x16x4_f32(false, av, false, bv,
                                                    (short)0, acc, false, false);
    }
    int r0 = tm * 16 + (lane >> 4) * 8;   // C/D: M = 8*group + vgpr
#pragma unroll
    for (int v = 0; v < 8; ++v)
        C[(r0 + v) * NOUT + ncol] = acc[v];
}

// ---------------------------------------------------------------------------
// K3: q[t] = sum_c W_gat[t,c] * att[128+c]    (1 block x 128)
// ---------------------------------------------------------------------------
__global__ void k_q(const float* __restrict__ Wg, const float* __restrict__ att,
                    float* __restrict__ q) {
    int t = threadIdx.x;
    float s = 0.0f;
    for (int c = 0; c < NOUT; ++c) s += Wg[t * NOUT + c] * att[NOUT + c];
    q[t] = s;
}

// ---------------------------------------------------------------------------
// K4: a[i] = xg[i] . att[:128]  (i<1536) ;  b[d] = heattr[d] . q  (d<1024)
//   grid = 20 blocks x 128  (2560 threads)
// ---------------------------------------------------------------------------
__global__ void k_ab(const float* __restrict__ xg, const float* __restrict__ att,
                     const float* __restrict__ heattr, const float* __restrict__ q,
                     float* __restrict__ a, float* __restrict__ bvec) {
    int id = blockIdx.x * blockDim.x + threadIdx.x;
    if (id < NCOMB) {
        float s = 0.0f;
        for (int c = 0; c < NOUT; ++c) s += xg[id * NOUT + c] * att[c];
        a[id] = s;
    } else {
        int d = id - NCOMB;   // 0..1023
        float s = 0.0f;
        for (int t = 0; t < NIN; ++t) s += heattr[d * NIN + t] * q[t];
        bvec[d] = s;
    }
}

// ---------------------------------------------------------------------------
// K5: per-edge softmax stats.  grid = 1024 blocks x 256 threads.
//   alpha[d] = softmax weight of the self (disease) member; also amax, Ssum,
//   wd1 = alpha^2, wd2 = alpha*(1-alpha)
// ---------------------------------------------------------------------------
__global__ __launch_bounds__(256) void k_edge(const float* __restrict__ a,
                                              const float* __restrict__ bvec,
                                              float* __restrict__ alpha,
                                              float* __restrict__ amax_o,
                                              float* __restrict__ ssum_o,
                                              float* __restrict__ wd1,
                                              float* __restrict__ wd2) {
    __shared__ float red[256];
    int d = blockIdx.x;
    int t = threadIdx.x;
    float bd = bvec[d];
    float lself = lrelu02(a[d] + bd);

    float mx = -3.4e38f;
    for (int m = t; m < MN; m += 256) mx = fmaxf(mx, lrelu02(a[DN + m] + bd));
    red[t] = mx; __syncthreads();
    for (int s = 128; s > 0; s >>= 1) { if (t < s) red[t] = fmaxf(red[t], red[t + s]); __syncthreads(); }
    float amax = fmaxf(red[0], lself);
    __syncthreads();

    float ss = 0.0f;
    for (int m = t; m < MN; m += 256) ss += expf(lrelu02(a[DN + m] + bd) - amax);
    red[t] = ss; __syncthreads();
    for (int s = 128; s > 0; s >>= 1) { if (t < s) red[t] += red[t + s]; __syncthreads(); }

    if (t == 0) {
        float es = expf(lself - amax);
        float S  = red[0] + es;
        float al = es / S;
        alpha[d] = al; amax_o[d] = amax; ssum_o[d] = S;
        wd1[d] = al * al;
        wd2[d] = al * (1.0f - al);
    }
}

// ---------------------------------------------------------------------------
// K6: per-medicine reductions over all 1024 edges.  grid = 2 x 256 (512 thr)
//   um1[m] = sum_d alpha[d]    * alpha(m,d)
//   um2[m] = sum_d (1-alpha[d])* alpha(m,d)
// ---------------------------------------------------------------------------
__global__ __launch_bounds__(256) void k_med(const float* __restrict__ a,
                                             const float* __restrict__ bvec,
                                             const float* __restrict__ alpha,
                                             const float* __restrict__ amax,
                                             const float* __restrict__ ssum,
                                             float* __restrict__ um1,
                                             float* __restrict__ um2) {
    int m = blockIdx.x * blockDim.x + threadIdx.x;   // 0..511
    float am = a[DN + m];
    float s1 = 0.0f, s2 = 0.0f;
    for (int d = 0; d < DN; ++d) {
        float al_md = expf(lrelu02(am + bvec[d]) - amax[d]) / ssum[d];
        float ad = alpha[d];
        s1 += ad * al_md;
        s2 += (1.0f - ad) * al_md;
    }
    um1[m] = s1; um2[m] = s2;
}

// ---------------------------------------------------------------------------
// K7: weighted column sums over xg + plain column sums over comb.
//   1 block x 128 threads; thread c owns column c (coalesced across threads).
// ---------------------------------------------------------------------------
__global__ void k_wsum(const float* __restrict__ xg, const float* __restrict__ comb,
                       const float* __restrict__ wd1, const float* __restrict__ wd2,
                       const float* __restrict__ um1, const float* __restrict__ um2,
                       float* __restrict__ t1, float* __restrict__ t2,
                       float* __restrict__ sdia, float* __restrict__ smed) {
    int c = threadIdx.x;
    float s1 = 0.0f, s2 = 0.0f, a1 = 0.0f, a2 = 0.0f;
    for (int d = 0; d < DN; ++d) {
        float x = xg[d * NOUT + c];
        s1 += wd1[d] * x;
        s2 += wd2[d] * x;
        a1 += comb[d * NIN + c];
    }
    for (int m = 0; m < MN; ++m) {
        float x = xg[(DN + m) * NOUT + c];
        s1 += um1[m] * x;
        s2 += um2[m] * x;
        a2 += comb[(DN + m) * NIN + c];
    }
    t1[c] = s1; t2[c] = s2; sdia[c] = a1; smed[c] = a2;
}

// ---------------------------------------------------------------------------
// K8: final fused matvecs.  1 block x 128 threads.
//   S1 = t1/513 + DN*b_gat ; S2 = t2/(513*DN) + MN*b_gat
//   v1 = (sdia/DN)@W_conv + b_conv ; v2 = (smed/MN)@W_conv + b_conv
//   i1 = S1@Wl[:128] + DN*v1@Wl[128:] ; i2 = S2@Wl[:128] + MN*v2@Wl[128:]
// ---------------------------------------------------------------------------
__global__ void k_final(const float* __restrict__ t1, const float* __restrict__ t2,
                        const float* __restrict__ sdia, const float* __restrict__ smed,
                        const float* __restrict__ Wc, const float* __restrict__ bc,
                        const float* __restrict__ bg, const float* __restrict__ Wl,
                        float* __restrict__ out) {
    __shared__ float S1[128], S2[128], V1[128], V2[128];
    int k = threadIdx.x;
    S1[k] = t1[k] * (1.0f / 513.0f) + (float)DN * bg[k];
    S2[k] = t2[k] * (1.0f / (513.0f * (float)DN)) + (float)MN * bg[k];
    float v1 = 0.0f, v2 = 0.0f;
    for (int t = 0; t < NIN; ++t) {
        float w = Wc[t * NOUT + k];
        v1 += sdia[t] * w;
        v2 += smed[t] * w;
    }
    V1[k] = v1 * (1.0f / (float)DN) + bc[k];
    V2[k] = v2 * (1.0f / (float)MN) + bc[k];
    __syncthreads();
    float o1 = 0.0f, o2 = 0.0f;
    for (int kk = 0; kk < 128; ++kk) {
        float wt = Wl[kk * NOUT + k];
        float wb = Wl[(128 + kk) * NOUT + k];
        o1 += S1[kk] * wt + (float)DN * V1[kk] * wb;
        o2 += S2[kk] * wt + (float)MN * V2[kk] * wb;
    }
    out[k]       = o1;
    out[128 + k] = o2;
}

// ---------------------------------------------------------------------------
extern "C" void kernel_launch(void* const* d_in, const int* in_sizes, int n_in,
                              void* d_out, int out_size, void* d_ws, size_t ws_size,
                              hipStream_t stream) {
    const int*   c_it   = (const int*)  d_in[0];
    const int*   m_it   = (const int*)  d_in[1];
    const float* cemb   = (const float*)d_in[2];
    const float* memb   = (const float*)d_in[3];
    const float* Wconv  = (const float*)d_in[4];
    const float* bconv  = (const float*)d_in[5];
    const float* Wgat   = (const float*)d_in[6];
    const float* bgat   = (const float*)d_in[7];
    const float* att    = (const float*)d_in[8];
    const float* heattr = (const float*)d_in[9];
    const float* Wlin   = (const float*)d_in[10];
    float* out = (float*)d_out;

    // workspace layout (floats)
    float* ws    = (float*)d_ws;
    float* comb  = ws;                 // 1536*128
    float* xg    = comb  + NCOMB*NIN;  // 1536*128
    float* q     = xg    + NCOMB*NOUT; // 128
    float* a     = q     + NIN;        // 1536
    float* bvec  = a     + NCOMB;      // 1024
    float* alpha = bvec  + DN;         // 1024
    float* amax  = alpha + DN;         // 1024
    float* ssum  = amax  + DN;         // 1024
    float* wd1   = ssum  + DN;         // 1024
    float* wd2   = wd1   + DN;         // 1024
    float* um1   = wd2   + DN;         // 512
    float* um2   = um1   + MN;         // 512
    float* t1    = um2   + MN;         // 128
    float* t2    = t1    + NOUT;       // 128
    float* sdia  = t2    + NOUT;       // 128
    float* smed  = sdia  + NIN;        // 128

    k_gather   <<<NCOMB, 128, 0, stream>>>(c_it, m_it, cemb, memb, comb);
    k_gemm_wmma<<<96,    256, 0, stream>>>(comb, Wgat, xg);
    k_q        <<<1,     128, 0, stream>>>(Wgat, att, q);
    k_ab       <<<20,    128, 0, stream>>>(xg, att, heattr, q, a, bvec);
    k_edge     <<<DN,    256, 0, stream>>>(a, bvec, alpha, amax, ssum, wd1, wd2);
    k_med      <<<2,     256, 0, stream>>>(a, bvec, alpha, amax, ssum, um1, um2);
    k_wsum     <<<1,     128, 0, stream>>>(xg, comb, wd1, wd2, um1, um2, t1, t2, sdia, smed);
    k_final    <<<1,     128, 0, stream>>>(t1, t2, sdia, smed, Wconv, bconv, bgat, Wlin, out);
}